// IntraClassVarianceLoss_9388798509066
// MI455X (gfx1250) — compile-verified
//
#include <hip/hip_runtime.h>
#include <hip/hip_bf16.h>

// Problem constants (match reference)
#define NCLS 100
#define NROW 131072
#define DDIM 512

// Workspace layout (floats)
#define ACC_FLOATS (NCLS * DDIM + 2 * NCLS) // 51400: sums[100][512], sq[100], cnt[100]
#define SQ_OFF     (NCLS * DDIM)            // 51200
#define CNT_OFF    (NCLS * DDIM + NCLS)     // 51300
#define PART_OFF   51456                    // 256B-aligned float offset for per-block partials

typedef __attribute__((ext_vector_type(2))) float v2f;
typedef __attribute__((ext_vector_type(8))) float v8f;

// ---------------------------------------------------------------------------
// Zero the global accumulator (needed only for the atomic-flush fallback).
// ---------------------------------------------------------------------------
__global__ void zero_acc_kernel(float* __restrict__ acc) {
    int i = blockIdx.x * blockDim.x + threadIdx.x;
    if (i < ACC_FLOATS) acc[i] = 0.0f;
}

// ---------------------------------------------------------------------------
// Bandwidth-dominant pass: stream features once, accumulate per-class
// sums / sq-norms / counts into a 205KB LDS accumulator (CDNA5: 320KB/WGP).
// One wave32 per row: float2 loads at lane-stride-32 -> coalesced 256B vmem
// bursts AND bank-conflict-free ds_add_f32 (banks 2*lane, 2*lane+1 distinct).
// ---------------------------------------------------------------------------
template <bool ATOMIC_FLUSH>
__global__ __launch_bounds__(1024, 1) void class_accum_kernel(
    const float* __restrict__ feat, const int* __restrict__ labels,
    float* __restrict__ acc, float* __restrict__ partials)
{
    __shared__ float lacc[ACC_FLOATS]; // 205,600 bytes LDS
    const int tid = threadIdx.x;

    for (int i = tid; i < ACC_FLOATS; i += 1024) lacc[i] = 0.0f;
    __syncthreads();

    const int lane = tid & 31;
    const int wid  = tid >> 5;
    const int gw   = blockIdx.x * 32 + wid; // global wave id
    const int nw   = gridDim.x * 32;        // total waves

    for (int row = gw; row < NROW; row += nw) {
        int lbl = labels[row];
        if ((unsigned)lbl >= (unsigned)NCLS) lbl = 0; // defensive clamp
        const v2f* __restrict__ rp = (const v2f*)(feat + (size_t)row * DDIM);
        float* __restrict__ cls = lacc + lbl * DDIM;

        float sq = 0.0f;
#pragma unroll
        for (int j = 0; j < 8; ++j) {
            v2f v = rp[lane + 32 * j]; // elements 2*(lane+32j), +1
            sq = __builtin_fmaf(v.x, v.x, sq);
            sq = __builtin_fmaf(v.y, v.y, sq);
            const int e = 2 * (lane + 32 * j);
            atomicAdd(&cls[e],     v.x); // ds_add_f32, bank 2*lane
            atomicAdd(&cls[e + 1], v.y); // ds_add_f32, bank 2*lane+1
        }
        // wave32 reduction of the squared-norm partials
#pragma unroll
        for (int off = 16; off > 0; off >>= 1)
            sq += __shfl_down(sq, off, 32);
        if (lane == 0) {
            atomicAdd(&lacc[SQ_OFF  + lbl], sq);
            atomicAdd(&lacc[CNT_OFF + lbl], 1.0f);
        }
    }
    __syncthreads();

    if (ATOMIC_FLUSH) {
        for (int i = tid; i < ACC_FLOATS; i += 1024)
            unsafeAtomicAdd(&acc[i], lacc[i]); // global_atomic_add_f32
    } else {
        float* __restrict__ slot = partials + (size_t)blockIdx.x * ACC_FLOATS;
        for (int i = tid; i < ACC_FLOATS; i += 1024)
            slot[i] = lacc[i]; // plain coalesced stores
    }
}

// ---------------------------------------------------------------------------
// Reduce per-block partial accumulators -> final accumulator (coalesced).
// ---------------------------------------------------------------------------
__global__ void reduce_partials_kernel(const float* __restrict__ partials,
                                       float* __restrict__ acc, int nslots)
{
    int i = blockIdx.x * blockDim.x + threadIdx.x;
    if (i >= ACC_FLOATS) return;
    float s = 0.0f;
    for (int k = 0; k < nslots; ++k)
        s += partials[(size_t)k * ACC_FLOATS + i];
    acc[i] = s;
}

// ---------------------------------------------------------------------------
// Finalize: per-class ||mu||^2 via v_wmma_f32_16x16x4_f32 as diag(mu @ mu^T).
// f32 A(16x4) and B(4x16) VGPR layouts are transposes of each other, so one
// float2 per lane (mu[m][2h], mu[m][2h+1]) serves BOTH operands. 8 waves:
// waves 0..6 each own a 16-class tile, accumulating 128 WMMA K-steps.
// ---------------------------------------------------------------------------
__global__ __launch_bounds__(256, 1) void finalize_loss_kernel(
    const float* __restrict__ acc, float* __restrict__ out)
{
    __shared__ float diag[7 * 256]; // one 16x16 D-tile per class-tile wave
    const int tid  = threadIdx.x;
    const int lane = tid & 31;
    const int w    = tid >> 5;

    if (w < 7) { // uniform per-wave branch: EXEC all-ones inside (WMMA req.)
        const int  half = lane >> 4;
        const int  m    = lane & 15;
        const int  cls  = 16 * w + m;
        const bool ok   = (cls < NCLS);
        const float cnt = ok ? acc[CNT_OFF + cls] : 0.0f;
        const float inv = (cnt > 0.0f) ? (1.0f / cnt) : 0.0f; // safe_n semantics
        const float* __restrict__ srow = acc + (size_t)(ok ? cls : 0) * DDIM;

        v8f c = {};
        for (int kb = 0; kb < DDIM; kb += 4) {
            const int k = kb + 2 * half;
            v2f ab;
            ab.x = srow[k]     * inv; // mu[m][k]   : A row m / B col m
            ab.y = srow[k + 1] * inv; // mu[m][k+1]
            c = __builtin_amdgcn_wmma_f32_16x16x4_f32(
                    false, ab, false, ab, (short)0, c, false, false);
        }
        // C/D layout: VGPR v -> M=v (lanes 0-15) or M=v+8 (lanes 16-31); N=m
        float* __restrict__ tile = diag + w * 256;
#pragma unroll
        for (int v = 0; v < 8; ++v)
            tile[(v + 8 * half) * 16 + m] = c[v];
    }
    __syncthreads();

    if (tid == 0) { // O(100) scalar epilogue
        float total = 0.0f, vc = 0.0f;
        for (int cc = 0; cc < NCLS; ++cc) {
            const float n    = acc[CNT_OFF + cc];
            const float musq = diag[(cc >> 4) * 256 + (cc & 15) * 16 + (cc & 15)];
            const float ssd  = acc[SQ_OFF + cc] - n * musq;
            const float tr   = ssd / fmaxf(n - 1.0f, 1.0f);
            if (n >= 2.0f) { total += tr; vc += 1.0f; }
        }
        out[0] = (vc > 0.0f) ? (total / vc) : 0.0f;
    }
}

// ---------------------------------------------------------------------------
extern "C" void kernel_launch(void* const* d_in, const int* in_sizes, int n_in,
                              void* d_out, int out_size, void* d_ws, size_t ws_size,
                              hipStream_t stream) {
    const float* feat   = (const float*)d_in[0];
    const int*   labels = (const int*)d_in[1];
    float*       out    = (float*)d_out;
    float*       ws     = (float*)d_ws;

    float* acc      = ws;            // ACC_FLOATS
    float* partials = ws + PART_OFF; // nslots * ACC_FLOATS

    const size_t avail = ws_size / sizeof(float);
    long slots = 0;
    if (avail > (size_t)PART_OFF)
        slots = (long)((avail - (size_t)PART_OFF) / (size_t)ACC_FLOATS);
    if (slots > 256) slots = 256;

    if (slots >= 64) {
        // Preferred: per-block partials + streaming serial per-element reduce (no atomics).
        const int gridB = (int)slots;
        class_accum_kernel<false><<<gridB, 1024, 0, stream>>>(feat, labels, acc, partials);
        reduce_partials_kernel<<<(ACC_FLOATS + 255) / 256, 256, 0, stream>>>(partials, acc, gridB);
    } else {
        // Fallback: global f32-atomic flush into a zeroed accumulator.
        zero_acc_kernel<<<(ACC_FLOATS + 255) / 256, 256, 0, stream>>>(acc);
        class_accum_kernel<true><<<256, 1024, 0, stream>>>(feat, labels, acc, partials);
    }

    finalize_loss_kernel<<<1, 256, 0, stream>>>(acc, out);
}